// RewardModel_85521388798036
// MI455X (gfx1250) — compile-verified
//
#include <hip/hip_runtime.h>
#include <hip/hip_bf16.h>
#include <math.h>

#define PAD_ID 0

typedef __attribute__((ext_vector_type(2))) float v2f;
typedef __attribute__((ext_vector_type(8))) float v8f;

// ---------------------------------------------------------------------------
// Kernel 1: per-batch index computation (one block per batch row).
//   start_ind = first s with mask!=0 (0 if none)
//   c_ind     = first s>=start with chosen_ids==PAD, else S
//   r_ind     = same for rejected if any divergence, else c_ind
//   last      = min(c_ind, r_ind) - 1, wrapped like a JAX negative index
// Traffic: 3 x 256KB int32 -> trivially HBM/L2 bound.
// ---------------------------------------------------------------------------
__global__ void rm_find_last(const int* __restrict__ ch_ids,
                             const int* __restrict__ ch_mask,
                             const int* __restrict__ rj_ids,
                             int* __restrict__ last_out,
                             int S) {
    const int b = blockIdx.x;
    __shared__ int s_start, s_div, s_c, s_r;
    if (threadIdx.x == 0) { s_start = S; s_div = 0; s_c = S; s_r = S; }
    __syncthreads();

    const int* ci = ch_ids  + (long long)b * S;
    const int* cm = ch_mask + (long long)b * S;
    const int* ri = rj_ids  + (long long)b * S;

    // Pass A: first nonzero mask position + divergence flag
    int lmin = S, ldiv = 0;
    for (int s = threadIdx.x; s < S; s += blockDim.x) {
        if (cm[s] != 0 && s < lmin) lmin = s;
        if (ci[s] != ri[s])         ldiv = 1;
    }
    atomicMin(&s_start, lmin);
    if (ldiv) atomicOr(&s_div, 1);
    __syncthreads();

    const int start = (s_start == S) ? 0 : s_start;  // argmax-of-all-false -> 0

    // Pass B: first PAD at/after start (ids re-read hit L2)
    int lc = S, lr = S;
    for (int s = threadIdx.x; s < S; s += blockDim.x) {
        if (s >= start) {
            if (ci[s] == PAD_ID && s < lc) lc = s;
            if (ri[s] == PAD_ID && s < lr) lr = s;
        }
    }
    atomicMin(&s_c, lc);
    atomicMin(&s_r, lr);
    __syncthreads();

    if (threadIdx.x == 0) {
        const int c_ind = s_c;
        const int r_ind = s_div ? s_r : c_ind;   // no-divergence branch: min(c_ind,S)==c_ind
        int last = min(c_ind, r_ind) - 1;
        if (last < 0) last += S;                 // JAX negative-index wrap semantics
        last_out[b] = last;
    }
}

// ---------------------------------------------------------------------------
// Kernel 2: one wave computes all 32 gathered dot products with
// V_WMMA_F32_16X16X4_F32.
//   A (16x4 f32): row m = hidden[m, last[m], k0:k0+4].
//     ISA layout: lanes 0-15 hold K=k0,k0+1 (VGPR0,1); lanes 16-31 hold k0+2,k0+3.
//   B (4x16 f32): every column = w[k0:k0+4] chunk (value depends only on K,
//     so the result is independent of the exact N-lane mapping).
//   D (16x16 f32): every column of row m = dot(hidden_row_m, w) -> after the
//     K loop, lane 0 holds m=0..7 in its 8 acc VGPRs, lane 16 holds m=8..15.
// Then those two lanes emit scores + the softplus loss reduction.
// ---------------------------------------------------------------------------
__global__ void rm_score_wmma(const float* __restrict__ ch_hid,
                              const float* __restrict__ rj_hid,
                              const float* __restrict__ w,
                              const int* __restrict__ last,
                              float* __restrict__ out,
                              int S, int H) {
    const int lane = threadIdx.x;           // 0..31, EXEC all ones (WMMA requirement)
    const int m    = lane & 15;             // A-matrix row = batch index
    const int ksub = (lane >> 4) * 2;       // K sub-offset within the 4-wide chunk

    const long long crow = ((long long)m * S + last[m]) * H;
    const long long rrow = ((long long)m * S + last[m]) * H;  // same last[] for both
    const float* cbase = ch_hid + crow;
    const float* rbase = rj_hid + rrow;

    v8f dc = {};  // chosen accumulator
    v8f dr = {};  // rejected accumulator

    #pragma unroll 4
    for (int k0 = 0; k0 < H; k0 += 4) {
        // 8-byte aligned: row base is H-aligned, k0 % 4 == 0, ksub in {0,2}
        const v2f a_c = *(const v2f*)(cbase + k0 + ksub);
        const v2f a_r = *(const v2f*)(rbase + k0 + ksub);
        const v2f bw  = *(const v2f*)(w     + k0 + ksub);
        dc = __builtin_amdgcn_wmma_f32_16x16x4_f32(false, a_c, false, bw,
                                                   (short)0, dc, false, false);
        dr = __builtin_amdgcn_wmma_f32_16x16x4_f32(false, a_r, false, bw,
                                                   (short)0, dr, false, false);
    }

    // Lane 0: rows m=0..7 ; lane 16: rows m=8..15 (C/D VGPR layout, 16x16 f32)
    __shared__ float part[2];
    if (lane == 0 || lane == 16) {
        const int base = (lane == 0) ? 0 : 8;
        float sum = 0.0f;
        #pragma unroll
        for (int r = 0; r < 8; ++r) {
            const float cs = dc[r];
            const float rs = dr[r];
            out[1  + base + r] = cs;          // chosen_mean_scores
            out[17 + base + r] = rs;          // rejected_mean_scores
            const float x = cs - rs;
            // -log_sigmoid(x) = softplus(-x), numerically stable form
            const float t = (x >= 0.0f) ? log1pf(expf(-x))
                                        : (-x + log1pf(expf(x)));
            sum += t;
        }
        part[lane >> 4] = sum;
    }
    __syncthreads();
    if (lane == 0) out[0] = (part[0] + part[1]) * (1.0f / 16.0f);  // mean loss
}

// ---------------------------------------------------------------------------
// Launch: derive shapes from sizes. Output = [loss(1) | chosen(16) | rej(16)].
// ---------------------------------------------------------------------------
extern "C" void kernel_launch(void* const* d_in, const int* in_sizes, int n_in,
                              void* d_out, int out_size, void* d_ws, size_t ws_size,
                              hipStream_t stream) {
    const int*   chosen_ids      = (const int*)d_in[0];
    const int*   chosen_mask     = (const int*)d_in[1];
    const int*   rejected_ids    = (const int*)d_in[2];
    const float* chosen_hidden   = (const float*)d_in[3];
    const float* rejected_hidden = (const float*)d_in[4];
    const float* v_head_w        = (const float*)d_in[5];
    float*       out             = (float*)d_out;

    const int B = (out_size - 1) / 2;   // 16
    const int S = in_sizes[0] / B;      // 4096
    const int H = in_sizes[5];          // 1024

    int* d_last = (int*)d_ws;

    rm_find_last<<<B, 256, 0, stream>>>(chosen_ids, chosen_mask, rejected_ids,
                                        d_last, S);
    rm_score_wmma<<<1, 32, 0, stream>>>(chosen_hidden, rejected_hidden, v_head_w,
                                        d_last, out, S, H);
}